// KAConvCComparision_78056735637895
// MI455X (gfx1250) — compile-verified
//
#include <hip/hip_runtime.h>

typedef __attribute__((ext_vector_type(2))) float v2f;
typedef __attribute__((ext_vector_type(8))) float v8f;

#define B_DIM 4
#define C_IN  16
#define H_DIM 64
#define W_DIM 64
#define OUT_C 32
#define M_TOT 144   // C_IN * 3 * 3
#define NUMD  6
#define DEND  4

__global__ __launch_bounds__(256) void ka_conv_wmma_f32(
    const float* __restrict__ in,      // [4,16,64,64]
    const float* __restrict__ nums,    // [32*144, 6]
    const float* __restrict__ denoms,  // [32*144, 4]
    float* __restrict__ out)           // [4,32,64,64]
{
    const int lane = threadIdx.x & 31;
    const int wave = threadIdx.x >> 5;
    const int tile = blockIdx.x * 8 + wave;      // 0..1023, one 16-pos tile per wave

    // 16 consecutive w positions in one (b, h) row (W=64 is a multiple of 16)
    const int pos0 = tile << 4;
    const int b    = pos0 >> 12;                 // / (64*64)
    const int h    = (pos0 >> 6) & 63;
    const int w0   = pos0 & 63;

    const int p    = lane & 15;                  // row (position) / col (out-ch) index
    const int half = lane >> 4;                  // lane-half: K = v + 2*half

    v8f acc0 = {0.f,0.f,0.f,0.f,0.f,0.f,0.f,0.f};
    v8f acc1 = {0.f,0.f,0.f,0.f,0.f,0.f,0.f,0.f};
    const v8f zero = {0.f,0.f,0.f,0.f,0.f,0.f,0.f,0.f};

    for (int m = 0; m < M_TOT; ++m) {
        const int c  = m / 9;
        const int kk = m - c * 9;
        const int di = kk / 3;
        const int dj = kk - di * 3;

        // patch value for position p (zero padding, branchless: clamp + select)
        const int hh = h + di - 1;
        const int ww = w0 + p + dj - 1;
        const bool valid = (hh >= 0) & (hh < H_DIM) & (ww >= 0) & (ww < W_DIM);
        const int hc = min(max(hh, 0), H_DIM - 1);
        const int wc = min(max(ww, 0), W_DIM - 1);
        float x = in[((b * C_IN + c) * H_DIM + hc) * W_DIM + wc];
        x = valid ? x : 0.0f;

        const float x2 = x * x, x3 = x2 * x, x4 = x2 * x2, x5 = x4 * x;

        // A fragments, 16x4 f32 layout: VGPR v, lane-half half -> K = v + 2*half
        const v2f a1 = half ? (v2f){x2, x3} : (v2f){1.0f, x};  // K=0..3 -> x^0..x^3
        const v2f a2 = half ? (v2f){0.f, 0.f} : (v2f){x4, x5}; // K=0,1 -> x^4,x^5; rest 0
        const v2f ad = half ? (v2f){x3, x4} : (v2f){x, x2};    // K -> x^(1+K)

        // ---- o-tile 0 (out channels 0..15) ----
        {
            const int o = p;
            const int nb = (o * M_TOT + m) * NUMD;
            const int db = (o * M_TOT + m) * DEND;
            // B fragments, 4x16 layout: VGPR v, lane-half half -> row K = v + 2*half
            const v2f b1 = { nums[nb + 2 * half + 0], nums[nb + 2 * half + 1] };
            const float n4 = nums[nb + 4], n5 = nums[nb + 5];        // always in-bounds
            const v2f b2 = { half ? 0.f : n4, half ? 0.f : n5 };
            const v2f bd = { denoms[db + 2 * half + 0], denoms[db + 2 * half + 1] };

            v8f num = __builtin_amdgcn_wmma_f32_16x16x4_f32(false, a1, false, b1,
                                                            (short)0, zero, false, false);
            num     = __builtin_amdgcn_wmma_f32_16x16x4_f32(false, a2, false, b2,
                                                            (short)0, num, false, false);
            v8f den = __builtin_amdgcn_wmma_f32_16x16x4_f32(false, ad, false, bd,
                                                            (short)0, zero, false, false);
            #pragma unroll
            for (int i = 0; i < 8; ++i) {
                const float q = 1.0f + __builtin_fabsf(den[i]);
                acc0[i] += num[i] * __builtin_amdgcn_rcpf(q);
            }
        }

        // ---- o-tile 1 (out channels 16..31) ----
        {
            const int o = 16 + p;
            const int nb = (o * M_TOT + m) * NUMD;
            const int db = (o * M_TOT + m) * DEND;
            const v2f b1 = { nums[nb + 2 * half + 0], nums[nb + 2 * half + 1] };
            const float n4 = nums[nb + 4], n5 = nums[nb + 5];
            const v2f b2 = { half ? 0.f : n4, half ? 0.f : n5 };
            const v2f bd = { denoms[db + 2 * half + 0], denoms[db + 2 * half + 1] };

            v8f num = __builtin_amdgcn_wmma_f32_16x16x4_f32(false, a1, false, b1,
                                                            (short)0, zero, false, false);
            num     = __builtin_amdgcn_wmma_f32_16x16x4_f32(false, a2, false, b2,
                                                            (short)0, num, false, false);
            v8f den = __builtin_amdgcn_wmma_f32_16x16x4_f32(false, ad, false, bd,
                                                            (short)0, zero, false, false);
            #pragma unroll
            for (int i = 0; i < 8; ++i) {
                const float q = 1.0f + __builtin_fabsf(den[i]);
                acc1[i] += num[i] * __builtin_amdgcn_rcpf(q);
            }
        }
    }

    // Store: C/D layout -> VGPR i, lane-half half holds (row = i + 8*half, col = p)
    const int oBase0 = ((b * OUT_C + p) * H_DIM + h) * W_DIM + w0;
    const int oBase1 = ((b * OUT_C + 16 + p) * H_DIM + h) * W_DIM + w0;
    #pragma unroll
    for (int i = 0; i < 8; ++i) {
        const int r = i + 8 * half;   // position within the 16-wide w tile
        out[oBase0 + r] = acc0[i];
        out[oBase1 + r] = acc1[i];
    }
}

extern "C" void kernel_launch(void* const* d_in, const int* in_sizes, int n_in,
                              void* d_out, int out_size, void* d_ws, size_t ws_size,
                              hipStream_t stream) {
    const float* in     = (const float*)d_in[0];   // [4,16,64,64]
    const float* nums   = (const float*)d_in[1];   // [4608, 6]
    const float* denoms = (const float*)d_in[2];   // [4608, 4]
    float* out          = (float*)d_out;           // [4,32,64,64]

    // 1024 waves total: 8 waves (256 threads) per block, 128 blocks
    ka_conv_wmma_f32<<<128, 256, 0, stream>>>(in, nums, denoms, out);
}